// MatrixKANLayer_14242111553787
// MI455X (gfx1250) — compile-verified
//
#include <hip/hip_runtime.h>
#include <hip/hip_bf16.h>

// MatrixKAN fused forward for MI455X (gfx1250, wave32).
// Store-bandwidth-bound (~1.6 GB of output per launch); spline contraction done
// in exact fp32 on the matrix unit via V_WMMA_F32_16X16X4_F32 (K=4 == spline
// window, 11 padded to 12 -> 3 WMMAs per (b-tile, o-tile, i)).

typedef __attribute__((ext_vector_type(2))) float v2f;
typedef __attribute__((ext_vector_type(8))) float v8f;

#define BATCH   2048
#define IN_DIM  256
#define OUT_DIM 256
#define NK      11      // NUM + K
#define IC      32      // i-chunk staged in LDS per pass
#define ICP     33      // padded LDS stride (bank-conflict avoidance)
#define SCP     17      // padded stride for [i][o] scale tiles

__device__ __forceinline__ float win_select(int d, float w0, float w1, float w2, float w3) {
    // full[n] = (0 <= n-idx <= 3) ? vals[n-idx] : 0   (d = n - idx)
    float r = 0.0f;
    r = (d == 0) ? w0 : r;
    r = (d == 1) ? w1 : r;
    r = (d == 2) ? w2 : r;
    r = (d == 3) ? w3 : r;
    return r;
}

__global__ __launch_bounds__(32) void kan_fused_kernel(
    const float* __restrict__ x,          // (B, I)
    const float* __restrict__ coef,       // (I, O, NK)
    const float* __restrict__ scale_base, // (I, O)
    const float* __restrict__ scale_sp,   // (I, O)
    const float* __restrict__ mask,       // (I, O)
    float* __restrict__ y,                // (B, O)
    float* __restrict__ preacts,          // (B, O, I)
    float* __restrict__ postacts,         // (B, O, I)
    float* __restrict__ postspline)       // (B, O, I)
{
    __shared__ float lds_spl[16 * 16 * ICP]; // [b_local*16 + o_local][i_local], stride ICP
    __shared__ float lds_x[16 * ICP];        // [b_local][i_local] staged inputs
    __shared__ float lds_xn[16 * ICP];
    __shared__ float lds_base[16 * ICP];
    __shared__ float lds_m[IC * SCP];        // [i_local][o_local] scale tiles
    __shared__ float lds_sb[IC * SCP];
    __shared__ float lds_ss[IC * SCP];

    const int lane = threadIdx.x;            // 0..31 (wave32)
    const int B0   = blockIdx.x * 16;
    const int O0   = blockIdx.y * 16;
    const int brow = lane & 15;              // A-matrix M row this lane feeds
    const int hi   = lane >> 4;              // half-wave: K={0,1} vs K={2,3}
    const int ocol = lane & 15;              // B/C/D N column

    float yacc[8];
#pragma unroll
    for (int r = 0; r < 8; ++r) yacc[r] = 0.0f;

    for (int ch = 0; ch < IN_DIM / IC; ++ch) {
        const int i0 = ch * IC;

        // ------------- Stage x rows and scale tiles for this chunk (coalesced) -------------
        for (int t = lane; t < 16 * IC; t += 32) {          // x: 16 b-rows x 32 i, 128B runs
            const int bl = t >> 5, il = t & 31;
            lds_x[bl * ICP + il] = x[(size_t)(B0 + bl) * IN_DIM + (i0 + il)];
        }
        for (int t = lane; t < IC * 16; t += 32) {          // scales: 32 i x 16 o
            const int il = t >> 4, ol = t & 15;
            const size_t po = (size_t)(i0 + il) * OUT_DIM + (O0 + ol);
            lds_m[il * SCP + ol]  = mask[po];
            lds_sb[il * SCP + ol] = scale_base[po];
            lds_ss[il * SCP + ol] = scale_sp[po];
        }
        __syncthreads();

        // ---------------- Phase A: per-i 16x16 spline GEMM via WMMA ----------------
        for (int il = 0; il < IC; ++il) {
            const int i = i0 + il;

            // per-(b,i) scalars (lanes 16-31 duplicate lanes 0-15's row math)
            float xv = lds_x[brow * ICP + il];
            // tanh(2x) = copysign(1 - 2e/(1+e), x), e = exp(-4|x|) in (0,1] (branch-free)
            float e4 = __expf(-4.0f * fabsf(xv));
            float xn = copysignf(1.0f - 2.0f * e4 * __builtin_amdgcn_rcpf(1.0f + e4), xv);
            // SiLU(xn) via v_rcp
            float bas = xn * __builtin_amdgcn_rcpf(1.0f + __expf(-xn));
            if (lane < 16) {
                lds_xn[brow * ICP + il]   = xn;
                lds_base[brow * ICP + il] = bas;
            }

            float pos  = (xn + 1.0f) * 4.0f;                 // (xn - GRID_LO)/h
            float fidx = fminf(fmaxf(floorf(pos), 0.0f), 7.0f);
            float u    = pos - fidx;
            int   idx  = (int)fidx;                          // window start in [0,7]
            float u2 = u * u, u3 = u2 * u;
            // cubic B-spline basis: [1 u u^2 u^3] @ (1/6)[[1,4,1,0],[-3,0,3,0],[3,-6,3,0],[-1,3,-3,1]]
            const float s6 = 1.0f / 6.0f;
            float w0 = s6 * (((-u + 3.0f) * u - 3.0f) * u + 1.0f); // (1-u)^3/6
            float w1 = s6 * ((3.0f * u - 6.0f) * u2 + 4.0f);
            float w2 = s6 * (((-3.0f * u + 3.0f) * u + 3.0f) * u + 1.0f);
            float w3 = s6 * u3;

            v8f acc = {};
            const float* cp = coef + ((size_t)i * OUT_DIM + (O0 + ocol)) * NK;
#pragma unroll
            for (int c = 0; c < 3; ++c) {                    // K chunks: n = 4c..4c+3 (11 padded to 12)
                const int n0 = c * 4 + hi * 2;
                // A fragment (16x4 f32): lane holds full[brow][n0], full[brow][n0+1]
                v2f a;
                a.x = win_select(n0 - idx,     w0, w1, w2, w3);
                a.y = win_select(n0 + 1 - idx, w0, w1, w2, w3);
                // B fragment (4x16 f32): rows K=n0,n0+1 at column ocol; pad n=11 with 0
                v2f b;
                b.x = cp[n0];
                b.y = (n0 + 1 < NK) ? cp[n0 + 1] : 0.0f;
                acc = __builtin_amdgcn_wmma_f32_16x16x4_f32(
                    /*neg_a=*/false, a, /*neg_b=*/false, b,
                    /*c_mod=*/(short)0, acc, /*reuse_a=*/false, /*reuse_b=*/false);
            }

            // y accumulation straight from the WMMA accumulator:
            // yacc[r] += m*(sb*base[b] + ss*spl[b,o]),  b = r + 8*hi, o = ocol
            const float m   = lds_m[il * SCP + ocol];
            const float msb = m * lds_sb[il * SCP + ocol];
            const float mss = m * lds_ss[il * SCP + ocol];
#pragma unroll
            for (int r = 0; r < 8; ++r) {
                float basr = __shfl(bas, r + 8 * hi, 32);    // ds_bpermute broadcast of base[b]
                yacc[r] = fmaf(mss, acc[r], fmaf(msb, basr, yacc[r]));
            }

            // stage spl tile into LDS transposed toward i-major
#pragma unroll
            for (int r = 0; r < 8; ++r) {
                const int bl = r + 8 * hi;                   // C layout: VGPR r -> M=r / M=r+8
                lds_spl[(bl * 16 + ocol) * ICP + il] = acc[r];
            }
        }
        __syncthreads();

        // ------------- Phase B: elementwise + coalesced i-major writeout -------------
#pragma unroll 4
        for (int row = 0; row < 256; ++row) {
            const int bl = row >> 4, ol = row & 15;
            const int i  = i0 + lane;                        // lane <-> i (contiguous stores)
            float spl = lds_spl[row * ICP + lane];
            float xnv = lds_xn[bl * ICP + lane];
            float bsv = lds_base[bl * ICP + lane];
            float m   = lds_m[lane * SCP + ol];
            float sb  = lds_sb[lane * SCP + ol];
            float ss  = lds_ss[lane * SCP + ol];
            float act = m * (sb * bsv + ss * spl);

            const size_t off = ((size_t)(B0 + bl) * OUT_DIM + (O0 + ol)) * IN_DIM + i;
            __builtin_nontemporal_store(xnv, preacts + off);    // preacts[b,o,i] = xn[b,i]
            __builtin_nontemporal_store(act, postacts + off);
            __builtin_nontemporal_store(spl, postspline + off); // raw spline (pre-mask/scale)
        }
        __syncthreads();
    }

    // flush y tile straight from registers (blocks own disjoint (b,o) tiles)
#pragma unroll
    for (int r = 0; r < 8; ++r) {
        y[(size_t)(B0 + r + 8 * hi) * OUT_DIM + (O0 + ocol)] = yacc[r];
    }
}

extern "C" void kernel_launch(void* const* d_in, const int* in_sizes, int n_in,
                              void* d_out, int out_size, void* d_ws, size_t ws_size,
                              hipStream_t stream) {
    const float* x          = (const float*)d_in[0];
    const float* coef       = (const float*)d_in[1];
    const float* scale_base = (const float*)d_in[2];
    const float* scale_sp   = (const float*)d_in[3];
    const float* mask       = (const float*)d_in[4];

    float* out        = (float*)d_out;
    float* y          = out;                                   // (B, O)
    float* preacts    = y + (size_t)BATCH * OUT_DIM;           // (B, O, I)
    float* postacts   = preacts + (size_t)BATCH * OUT_DIM * IN_DIM;
    float* postspline = postacts + (size_t)BATCH * OUT_DIM * IN_DIM;

    dim3 grid(BATCH / 16, OUT_DIM / 16);                       // 2048 single-wave blocks
    kan_fused_kernel<<<grid, 32, 0, stream>>>(x, coef, scale_base, scale_sp, mask,
                                              y, preacts, postacts, postspline);
}